// LSTM_13451837571407
// MI455X (gfx1250) — compile-verified
//
#include <hip/hip_runtime.h>
#include <hip/hip_bf16.h>

typedef __attribute__((ext_vector_type(2))) float v2f;
typedef __attribute__((ext_vector_type(8))) float v8f;

#define LSTM_B 1048576
#define LSTM_I 14
#define LSTM_H 16
#define LSTM_L 8
#define NTILES (LSTM_B / 16)            // 65536 tiles of 16 batch rows
#define WS_W_FLOATS (LSTM_L * 16 * 64)  // 8192 floats, fragment-swizzled
#define WS_B_OFF WS_W_FLOATS            // 8192: bias[l][n] (512)
#define WS_FC_OFF (WS_B_OFF + LSTM_L * 64)  // 8704: fcw[16], fcb
#define WS_TOTAL (WS_FC_OFF + 17)       // 8721 floats

// fast sigmoid / tanh built on v_exp_f32 + raw v_rcp_f32 (no div fixup)
__device__ __forceinline__ float fast_sigmoid(float x) {
  return __builtin_amdgcn_rcpf(1.0f + __expf(-x));
}
__device__ __forceinline__ float fast_tanh(float x) {
  return __builtin_fmaf(2.0f, __builtin_amdgcn_rcpf(1.0f + __expf(-2.0f * x)),
                        -1.0f);
}

// ---------------------------------------------------------------------------
// Prep: fold per-layer (W_ih + W_hh) and store in per-lane WMMA B-fragment
// order:  frag[l][j][t][lane][comp]  (j = K-chunk of 4, t = N-tile of 16)
//   k = 4j + 2*(lane>=16) + comp ,  n = 16t + lane%16
// so the main kernel loads each B operand with ONE 8B-aligned ds_load_b64.
// Layer 0 uses W_ih0 (K padded 14->16 with zeros); its W_hh term vanishes
// (h0 == 0) but b_hh[0] is kept in the folded bias.
// ---------------------------------------------------------------------------
__global__ void lstm_prep_kernel(const float* __restrict__ Wih0,
                                 const float* __restrict__ Wih,
                                 const float* __restrict__ Whh,
                                 const float* __restrict__ bih,
                                 const float* __restrict__ bhh,
                                 const float* __restrict__ fcw,
                                 const float* __restrict__ fcb,
                                 float* __restrict__ ws) {
  int idx = blockIdx.x * 256 + threadIdx.x;
  if (idx < WS_W_FLOATS) {
    int comp = idx & 1;
    int lane = (idx >> 1) & 31;
    int t    = (idx >> 6) & 3;
    int j    = (idx >> 8) & 3;
    int l    = idx >> 10;
    int k = 4 * j + 2 * (lane >> 4) + comp;
    int n = 16 * t + (lane & 15);
    float v;
    if (l == 0) {
      v = (k < LSTM_I) ? Wih0[n * LSTM_I + k] : 0.0f;
    } else {
      v = Wih[((l - 1) * 64 + n) * 16 + k] + Whh[(l * 64 + n) * 16 + k];
    }
    ws[idx] = v;
  } else if (idx < WS_FC_OFF) {
    int i = idx - WS_B_OFF;
    ws[idx] = bih[i] + bhh[i];
  } else if (idx < WS_FC_OFF + 16) {
    ws[idx] = fcw[idx - WS_FC_OFF];
  } else if (idx == WS_FC_OFF + 16) {
    ws[idx] = fcb[0];
  }
}

// ---------------------------------------------------------------------------
// Main fused LSTM kernel. One wave handles a 16-row batch tile end to end.
// G[16,64] = H[16,16] @ W[l][16,64] via 4x4 chained v_wmma_f32_16x16x4_f32.
// ---------------------------------------------------------------------------
#define STG_STRIDE 18  // padded row stride (floats): conflict-free, 8B aligned

__global__ __launch_bounds__(256) void lstm_main_kernel(
    const float* __restrict__ x, const float* __restrict__ ws,
    float* __restrict__ out, int totalWaves) {
  __shared__ float sW[WS_W_FLOATS];      // 32 KB weights, fragment-swizzled
  __shared__ float sB[LSTM_L * 64];      // folded biases
  __shared__ float sFc[17];              // fc weights + bias
  __shared__ float sStage[8][16 * STG_STRIDE];  // per-wave 16x16 staging

  const int tid = threadIdx.x;
  for (int i = tid; i < WS_TOTAL; i += 256) {
    float v = ws[i];
    if (i < WS_W_FLOATS)      sW[i] = v;
    else if (i < WS_FC_OFF)   sB[i - WS_B_OFF] = v;
    else                      sFc[i - WS_FC_OFF] = v;
  }
  __syncthreads();

  const int lane = tid & 31;
  const int warp = tid >> 5;
  const int half = lane >> 4;   // 0: lanes 0-15, 1: lanes 16-31
  const int l16  = lane & 15;
  float* stg = sStage[warp];

  const int gw = blockIdx.x * 8 + warp;

  for (int tile = gw; tile < NTILES; tile += totalWaves) {
    // ---- stage x tile (16 rows x 14 floats = 224 consecutive dwords) ----
    const float* xt = x + (size_t)tile * (16 * LSTM_I);
#pragma unroll
    for (int it = 0; it < 7; ++it) {
      int f = lane + it * 32;            // 0..223
      float v = xt[f];
      int r = f / LSTM_I, c = f % LSTM_I;
      stg[r * STG_STRIDE + c] = v;
    }
    if (half == 0) {  // zero-pad K = 14,15
      stg[l16 * STG_STRIDE + 14] = 0.0f;
      stg[l16 * STG_STRIDE + 15] = 0.0f;
    }
    asm volatile("s_wait_dscnt 0" ::: "memory");

    v8f c8 = {0.f, 0.f, 0.f, 0.f, 0.f, 0.f, 0.f, 0.f};
    v8f h8;

#pragma unroll
    for (int l = 0; l < LSTM_L; ++l) {
      // C init = folded bias (n = lane%16 is constant across the 8 vgprs)
      v8f acc[4];
#pragma unroll
      for (int t = 0; t < 4; ++t) {
        float bv = sB[l * 64 + t * 16 + l16];
        acc[t] = (v8f){bv, bv, bv, bv, bv, bv, bv, bv};
      }
      // K = 16 as 4 chained K=4 WMMAs; N = 64 as 4 column tiles (i,f,g,o)
#pragma unroll
      for (int j = 0; j < 4; ++j) {
        const int k0 = 4 * j + 2 * half;  // A layout: vgpr0/1 = K+0/K+1,
        v2f a;                            // lane half selects K+2 offset
        a.x = stg[l16 * STG_STRIDE + k0];
        a.y = stg[l16 * STG_STRIDE + k0 + 1];
#pragma unroll
        for (int t = 0; t < 4; ++t) {
          // one 8B ds_load_b64 per B fragment (pre-swizzled, conflict-free)
          v2f b = *(const v2f*)&sW[(((l * 4 + j) * 4 + t) * 32 + lane) * 2];
          acc[t] = __builtin_amdgcn_wmma_f32_16x16x4_f32(
              false, a, false, b, (short)0, acc[t], false, false);
        }
      }
      // ---- LSTM cell, elementwise in C-fragment layout ----
#pragma unroll
      for (int r = 0; r < 8; ++r) {
        float si = fast_sigmoid(acc[0][r]);
        float sf = fast_sigmoid(acc[1][r]);
        float tg = fast_tanh(acc[2][r]);
        float so = fast_sigmoid(acc[3][r]);
        float cn = __builtin_fmaf(sf, c8[r], si * tg);
        c8[r] = cn;
        h8[r] = so * fast_tanh(cn);
      }
      // ---- transpose h: C layout -> row-major staging (next layer's A) ----
      asm volatile("" ::: "memory");  // keep LDS reads above the stores
#pragma unroll
      for (int r = 0; r < 8; ++r) {
        int m = r + 8 * half;
        stg[m * STG_STRIDE + l16] = h8[r];
      }
      asm volatile("s_wait_dscnt 0" ::: "memory");
    }

    // ---- final FC: lane m (<16) reduces staged row m ----
    if (half == 0) {
      float s = sFc[16];
#pragma unroll
      for (int n = 0; n < 16; ++n)
        s = __builtin_fmaf(stg[l16 * STG_STRIDE + n], sFc[n], s);
      out[(size_t)tile * 16 + l16] = s;
    }
  }
}

// ---------------------------------------------------------------------------
extern "C" void kernel_launch(void* const* d_in, const int* in_sizes, int n_in,
                              void* d_out, int out_size, void* d_ws,
                              size_t ws_size, hipStream_t stream) {
  const float* x    = (const float*)d_in[0];
  const float* Wih0 = (const float*)d_in[1];
  const float* Wih  = (const float*)d_in[2];
  const float* Whh  = (const float*)d_in[3];
  const float* bih  = (const float*)d_in[4];
  const float* bhh  = (const float*)d_in[5];
  const float* fcw  = (const float*)d_in[6];
  const float* fcb  = (const float*)d_in[7];
  float* out = (float*)d_out;
  float* ws  = (float*)d_ws;

  lstm_prep_kernel<<<(WS_TOTAL + 255) / 256, 256, 0, stream>>>(
      Wih0, Wih, Whh, bih, bhh, fcw, fcb, ws);

  const int blocks = 2048;               // 16384 waves -> 4 tiles each, uniform
  const int totalWaves = blocks * 8;
  lstm_main_kernel<<<blocks, 256, 0, stream>>>(x, ws, out, totalWaves);
}